// PGExplainer_63806034149709
// MI455X (gfx1250) — compile-verified
//
#include <hip/hip_runtime.h>

typedef __attribute__((ext_vector_type(2))) float v2f;
typedef __attribute__((ext_vector_type(8))) float v8f;

#define NN 2048
#define FF 128
#define HH 64
#define EE 65536

// -------------------------------------------------------------------------
// Kernel 1: G1 = embed @ W1[0:128,:],  G2 = embed @ W1[128:256,:]
// via V_WMMA_F32_16X16X4_F32.  One wave computes one 16x16 tile of one G.
// 1024 waves total: waves [0,512) -> G1 tiles, [512,1024) -> G2 tiles.
// -------------------------------------------------------------------------
__global__ __launch_bounds__(256) void pgx_gemm_wmma(
    const float* __restrict__ embed,   // [2048,128] row major
    const float* __restrict__ W1,      // [256,64]   row major
    float* __restrict__ g1,            // [2048,64]
    float* __restrict__ g2)            // [2048,64]
{
  const int lane = threadIdx.x & 31;
  const int wave = (blockIdx.x * blockDim.x + threadIdx.x) >> 5;
  const int which = wave >> 9;        // 0 -> g1, 1 -> g2
  const int tile  = wave & 511;
  const int ti = tile >> 2;           // row tile 0..127
  const int tj = tile & 3;            // col tile 0..3

  const int m   = (lane & 15);        // matrix row (A) / col (B,C,D) index
  const int row = ti * 16 + m;        // A-matrix row in embed
  const int col = tj * 16 + m;        // B/D column
  const int khalf = (lane >> 4) << 1; // 0 for lanes 0-15, 2 for lanes 16-31

  const float* __restrict__ w1base = W1 + which * (FF * HH);

  v8f acc = {0.f, 0.f, 0.f, 0.f, 0.f, 0.f, 0.f, 0.f};

#pragma unroll 4
  for (int k0 = 0; k0 < FF; k0 += 4) {
    // A fragment 16x4 (ISA 7.12.2): lane L<16 holds K=k0,k0+1; L>=16 holds K=k0+2,k0+3
    v2f a;
    a.x = embed[row * FF + k0 + khalf];
    a.y = embed[row * FF + k0 + khalf + 1];
    // B fragment 4x16: row striped across lanes within a VGPR
    v2f b;
    b.x = w1base[(k0 + khalf) * HH + col];
    b.y = w1base[(k0 + khalf + 1) * HH + col];
    // 8 args: (neg_a, A, neg_b, B, c_mod, C, reuse_a, reuse_b)
    acc = __builtin_amdgcn_wmma_f32_16x16x4_f32(
        false, a, false, b, (short)0, acc, false, false);
  }

  float* __restrict__ G = which ? g2 : g1;
  const int rbase = ti * 16 + ((lane >> 4) << 3);  // lanes 16-31 hold rows M=8..15
#pragma unroll
  for (int r = 0; r < 8; ++r)
    G[(rbase + r) * HH + col] = acc[r];
}

// -------------------------------------------------------------------------
// Kernel 2: per-edge concrete mask.  Evaluates log_alpha only at (i,j) and
// (j,i) instead of materializing the full NxN matrix (32x less work).
// -------------------------------------------------------------------------
__global__ __launch_bounds__(256) void pgx_edge_mask(
    const float* __restrict__ g1, const float* __restrict__ g2,
    const float* __restrict__ b1, const float* __restrict__ W2,
    const float* __restrict__ b2,
    const int*   __restrict__ edge_index,   // [2,65536]
    const float* __restrict__ noise,        // [2048,2048]
    float* __restrict__ out)                // [65536]
{
  const int e = blockIdx.x * blockDim.x + threadIdx.x;
  if (e >= EE) return;

  const int i = edge_index[e];
  const int j = edge_index[EE + e];

  const float* __restrict__ g1i = g1 + i * HH;
  const float* __restrict__ g2i = g2 + i * HH;
  const float* __restrict__ g1j = g1 + j * HH;
  const float* __restrict__ g2j = g2 + j * HH;

  float la1 = 0.f;   // log_alpha(i,j)
  float la2 = 0.f;   // log_alpha(j,i)
#pragma unroll 8
  for (int h = 0; h < HH; ++h) {
    const float w  = W2[h];   // wave-uniform -> scalar load
    const float bb = b1[h];   // wave-uniform -> scalar load
    la1 += fmaxf(g1i[h] + g2j[h] + bb, 0.f) * w;
    la2 += fmaxf(g1j[h] + g2i[h] + bb, 0.f) * w;
  }
  const float bias2 = b2[0];
  la1 += bias2;
  la2 += bias2;

  const float uij = noise[(long)i * NN + j];
  const float uji = noise[(long)j * NN + i];
  const float lij = __logf(uij) - log1pf(-uij);
  const float lji = __logf(uji) - log1pf(-uji);

  // BETA == 1.0
  const float m1 = 1.f / (1.f + __expf(-(lij + la1)));
  const float m2 = 1.f / (1.f + __expf(-(lji + la2)));
  out[e] = 0.5f * (m1 + m2);
}

extern "C" void kernel_launch(void* const* d_in, const int* in_sizes, int n_in,
                              void* d_out, int out_size, void* d_ws, size_t ws_size,
                              hipStream_t stream) {
  const float* embed      = (const float*)d_in[0];
  const int*   edge_index = (const int*)  d_in[1];
  const float* noise      = (const float*)d_in[2];
  const float* W1         = (const float*)d_in[3];
  const float* b1         = (const float*)d_in[4];
  const float* W2         = (const float*)d_in[5];
  const float* b2         = (const float*)d_in[6];

  float* g1 = (float*)d_ws;          // [2048*64]
  float* g2 = g1 + NN * HH;          // [2048*64]   (1 MB total workspace)

  // 1024 waves = 32768 threads; 128 blocks x 256 threads
  pgx_gemm_wmma<<<128, 256, 0, stream>>>(embed, W1, g1, g2);
  pgx_edge_mask<<<EE / 256, 256, 0, stream>>>(g1, g2, b1, W2, b2,
                                              edge_index, noise, (float*)d_out);
}